// AttentionDownsampler_10264971837445
// MI455X (gfx1250) — compile-verified
//
#include <hip/hip_runtime.h>

// AttentionDownsampler for MI455X (gfx1250).
// One workgroup (256 threads / 8 wave32) per output position (b,hh,ww).
// Patch (14x14x384 f32, 301KB) staged into LDS by the Tensor Data Mover
// (4 descriptors, one per wave for waves 0..3), then:
//   phase1: logits via V_WMMA_F32_16X16X4_F32   (A = attn_w, B = patch^T)
//   softmax: wave32 shuffle reduction
//   phase2: weighted channel reduction via V_WMMA_F32_16X16X4_F32
#define KSZ    14
#define NPATCH 196          // 14*14
#define PPAD   196          // TDM writes rows contiguously; even -> b64 aligned
#define CDIM   384
#define HOUT   16
#define WOUT   16

typedef __attribute__((ext_vector_type(2))) float v2f;
typedef __attribute__((ext_vector_type(8))) float v8f;
typedef __attribute__((ext_vector_type(4))) unsigned int v4u;
typedef __attribute__((ext_vector_type(8))) int v8i;
typedef __attribute__((ext_vector_type(4))) int v4i;

__launch_bounds__(256)
__global__ void attn_downsample_kernel(
    const float* __restrict__ hr,            // (4,384,224,224)
    const float* __restrict__ attn_w,        // (1,384)
    const float* __restrict__ attn_b,        // (1,)
    const float* __restrict__ w_kk,          // (14,14) flattened over p
    const float* __restrict__ b_kk,          // (14,14)
    const unsigned char* __restrict__ dmask, // (4,16,16,1) bool
    float* __restrict__ out)                 // (4,384,16,16)
{
    extern __shared__ float smem[];
    float* sh_attn  = smem;          // 208 floats (logits -> attn)
    float* sh_red   = smem + 208;    // 8 floats (max, 1/sum)
    float* sh_w     = smem + 216;    // 384 floats (attn_w)
    float* sh_patch = smem + 600;    // 385 * PPAD floats, layout [c][p]

    const int t    = threadIdx.x;
    const int lane = t & 31;
    const int wid  = t >> 5;        // wave id 0..7
    const int nl   = lane & 15;     // N index within fragment
    const int half = lane >> 4;     // 0: K=0,1  1: K=2,3

    const int blk = blockIdx.x;     // b*256 + hh*16 + ww
    const int b   = blk >> 8;
    const int hh  = (blk >> 4) & 15;
    const int ww  = blk & 15;

    // global corner of this patch (channel 0)
    const float* src = hr + (((size_t)b * CDIM) * 224 + (size_t)hh * KSZ) * 224
                          + (size_t)ww * KSZ;

    // ---------------- Phase 0: TDM-stage patch into LDS ---------------------
    // Waves 0..3 each issue one tensor_load_to_lds for a 14x14x96 f32 tile:
    //   dim0 = 14 elems (row), stride0 = 224; dim1 = 14 rows, stride1 = 50176
    //   (channel pitch); dim2 = 96 channels. LDS dest is contiguous [c][196].
    if (wid < 4) {
        const unsigned long long ga =
            (unsigned long long)(const void*)src +
            (unsigned long long)wid * 96ull * 50176ull * 4ull;
        const unsigned int lds_byte = (600u + (unsigned)wid * 96u * PPAD) * 4u;
        // D# group 0: count=1 | lds_addr | global_addr[56:0] | type=2
        v4u g0 = { 1u,
                   lds_byte,
                   (unsigned)(ga & 0xFFFFFFFFu),
                   ((unsigned)(ga >> 32) & 0x01FFFFFFu) | 0x80000000u };
        // D# group 1: data_size=4B; tensor_dim0/1=224; tile=(14,14,96);
        //             stride0=224; stride1=50176
        v8i g1 = { (int)0x00020000,          // data_size=2 (4B)
                   (int)(224u << 16),        // tensor_dim0[15:0] in [31:16]
                   (int)(224u << 16),        // dim0 hi=0 | tensor_dim1[15:0]
                   (int)(14u  << 16),        // dim1 hi=0 | tile_dim0=14
                   (int)(14u | (96u << 16)), // tile_dim1=14 | tile_dim2=96
                   (int)224,                 // tensor_dim0_stride[31:0]
                   (int)0xC4000000u,         // stride0 hi=0 | stride1 lo=50176
                   0 };                      // stride1 hi
        v4i g2 = { (int)0x00FFFFFF, 0, 0, 0 }; // tensor_dim2 large; no dim3
        v4i g3 = { 0, 0, 0, 0 };
        v8i gx = { 0, 0, 0, 0, 0, 0, 0, 0 };   // unused trailing group (zero)
        __builtin_amdgcn_tensor_load_to_lds(g0, g1, g2, g3, gx, 0);
        __builtin_amdgcn_s_wait_tensorcnt(0);
    }
    sh_w[t] = attn_w[t];
    if (t < CDIM - 256) sh_w[256 + t] = attn_w[256 + t];
    __syncthreads();

    const float ab = attn_b[0];
    const float mk = dmask[((b * HOUT) + hh) * WOUT + ww] ? 1.0f : 0.0f;
    const bool amask = (nl == 0);   // only M=0 row of the A fragment is live

    // ---------------- Phase 1: logits[p] = sum_c patch[p][c]*w[c] -----------
    // D[0][n] = sum_k w[c0+k] * patch[c0+k][p0+n]; accumulate over all c.
    for (int tile = wid; tile < 13; tile += 8) {
        const int p0 = tile << 4;
        v8f acc = {};
        for (int cc = 0; cc < CDIM; cc += 4) {
            const float w0 = sh_w[cc + 2 * half];       // broadcast reads
            const float w1 = sh_w[cc + 2 * half + 1];
            v2f a;
            a[0] = amask ? w0 : 0.0f;                   // v_cndmask, no EXEC ops
            a[1] = amask ? w1 : 0.0f;
            v2f bf;
            bf[0] = sh_patch[(cc + 2 * half) * PPAD + p0 + nl];
            bf[1] = sh_patch[(cc + 2 * half + 1) * PPAD + p0 + nl];
            acc = __builtin_amdgcn_wmma_f32_16x16x4_f32(
                      false, a, false, bf, (short)0, acc, false, false);
        }
        const int p = p0 + nl;
        if (half == 0 && p < NPATCH) {
            // logits = (dot + attn_b) * mask * w_kk + b_kk
            sh_attn[p] = (acc[0] + ab) * mk * w_kk[p] + b_kk[p];
        }
    }
    __syncthreads();

    // ---------------- Softmax over p (wave32 shuffle reduction) -------------
    if (t < 32) {
        float m = -3.4e38f;
        for (int p = lane; p < NPATCH; p += 32) m = fmaxf(m, sh_attn[p]);
        for (int off = 16; off; off >>= 1) m = fmaxf(m, __shfl_xor(m, off, 32));
        float s = 0.0f;
        for (int p = lane; p < NPATCH; p += 32) s += __expf(sh_attn[p] - m);
        for (int off = 16; off; off >>= 1) s += __shfl_xor(s, off, 32);
        if (lane == 0) { sh_red[0] = m; sh_red[1] = 1.0f / s; }
    }
    __syncthreads();
    {
        const float m = sh_red[0], inv = sh_red[1];
        if (t < NPATCH) sh_attn[t] = __expf(sh_attn[t] - m) * inv;
    }
    __syncthreads();

    // ---------------- Phase 2: out[c] = sum_p attn[p]*patch[p][c] -----------
    // D[0][n] = sum_k attn[p0+k] * patch[p0+k][c0+n]; 24 c-tiles, 49 K-steps.
    for (int tile = wid; tile < 24; tile += 8) {
        const int c0 = tile << 4;
        v8f acc = {};
        for (int p0 = 0; p0 < NPATCH; p0 += 4) {
            const float a0 = sh_attn[p0 + 2 * half];    // broadcast reads
            const float a1 = sh_attn[p0 + 2 * half + 1];
            v2f a;
            a[0] = amask ? a0 : 0.0f;
            a[1] = amask ? a1 : 0.0f;
            // B[k][n] = patch[p0+k][c0+n]; contiguous pair -> ds_load_b64
            v2f bf = *(const v2f*)(sh_patch + (c0 + nl) * PPAD + p0 + 2 * half);
            acc = __builtin_amdgcn_wmma_f32_16x16x4_f32(
                      false, a, false, bf, (short)0, acc, false, false);
        }
        if (half == 0) {
            const int c = c0 + nl;
            out[(((size_t)b * CDIM + c) * HOUT + hh) * WOUT + ww] = acc[0];
        }
    }
}

extern "C" void kernel_launch(void* const* d_in, const int* in_sizes, int n_in,
                              void* d_out, int out_size, void* d_ws, size_t ws_size,
                              hipStream_t stream) {
    // setup_inputs order: hr_feats, guidance(unused), attn_w, attn_b, w_kk, b_kk, dropout_mask
    const float* hr      = (const float*)d_in[0];
    const float* attn_w  = (const float*)d_in[2];
    const float* attn_b  = (const float*)d_in[3];
    const float* w_kk    = (const float*)d_in[4];
    const float* b_kk    = (const float*)d_in[5];
    const unsigned char* dmask = (const unsigned char*)d_in[6];
    float* out = (float*)d_out;

    const size_t shbytes = (size_t)(600 + 385 * PPAD) * sizeof(float); // ~304 KB
    attn_downsample_kernel<<<dim3(4 * HOUT * WOUT), dim3(256), shbytes, stream>>>(
        hr, attn_w, attn_b, w_kk, b_kk, dmask, out);
}